// Attention_86371792322733
// MI455X (gfx1250) — compile-verified
//
#include <hip/hip_runtime.h>
#include <hip/hip_bf16.h>
#include <math.h>

typedef float v2f __attribute__((ext_vector_type(2)));
typedef float v8f __attribute__((ext_vector_type(8)));

#define WMMA_F32(a, b, c) \
  __builtin_amdgcn_wmma_f32_16x16x4_f32(false, (a), false, (b), (short)0, (c), false, false)

constexpr int Bb = 4, Ss = 1024, Dd = 768, Hh = 12, HDh = 64;

// ---------------------------------------------------------------------------
// Generic fp32 WMMA GEMM: C[M,N] = alpha * op(A)[M,K] @ op(B)[K,N] + bias[N]
// Block tile 64x128, K tile 32, 4 waves (128 threads), each wave 32x64
// (2x4 16x16 frags -> 8 WMMAs per 6 LDS pair-loads per K=4 step).
// As is m-major [m][k]; Bs is n-major [n][k] so both fragment dword-pairs are
// k-contiguous in LDS (single ds_load_2addr, no repacking moves).
// Call sites: M % 64 == 0, N % 128 == 0, K % 32 == 0 (no edge guards needed).
// ---------------------------------------------------------------------------
__global__ __launch_bounds__(128)
void gemm_wmma_f32(const float* __restrict__ A, const float* __restrict__ Bm,
                   float* __restrict__ C, const float* __restrict__ bias,
                   int M, int N, int K, int lda, int ldb, int ldc,
                   float alpha, int transA, int transB) {
  __shared__ float As[64][33];    // [m][k], +1 pad
  __shared__ float Bs[128][33];   // [n][k], +1 pad
  const int tid  = threadIdx.x;
  const int lane = tid & 31;
  const int wid  = tid >> 5;
  const int wm = wid >> 1, wn = wid & 1;   // wave tile: rows wm*32, cols wn*64
  const int m0 = blockIdx.y * 64;
  const int n0 = blockIdx.x * 128;
  const int l15 = lane & 15;
  const int kh  = (lane >> 4) << 1;   // 0 for lanes 0-15, 2 for lanes 16-31

  v8f acc[2][4] = {};

  for (int k0 = 0; k0 < K; k0 += 32) {
    if (!transA) {
      for (int t = tid; t < 64 * 32; t += 128) {       // consecutive t -> consecutive k
        int r = t >> 5, c = t & 31;
        As[r][c] = A[(size_t)(m0 + r) * lda + (k0 + c)];
      }
    } else {
      for (int t = tid; t < 64 * 32; t += 128) {       // consecutive t -> consecutive m
        int r = t & 63, c = t >> 6;
        As[r][c] = A[(size_t)(k0 + c) * lda + (m0 + r)];
      }
    }
    if (!transB) {
      for (int t = tid; t < 128 * 32; t += 128) {      // consecutive t -> consecutive n
        int c = t & 127, r = t >> 7;
        Bs[c][r] = Bm[(size_t)(k0 + r) * ldb + (n0 + c)];
      }
    } else {
      for (int t = tid; t < 128 * 32; t += 128) {      // consecutive t -> consecutive k
        int r = t & 31, c = t >> 5;
        Bs[c][r] = Bm[(size_t)(n0 + c) * ldb + (k0 + r)];
      }
    }
    __syncthreads();
#pragma unroll
    for (int kk = 0; kk < 32; kk += 4) {
      v2f a[2], b[4];
#pragma unroll
      for (int i = 0; i < 2; ++i) {
        int row = wm * 32 + i * 16 + l15;          // A frag: M = lane%16 (+tile)
        a[i].x = As[row][kk + kh];                  // VGPR0: K = kh
        a[i].y = As[row][kk + kh + 1];              // VGPR1: K = kh+1
      }
#pragma unroll
      for (int j = 0; j < 4; ++j) {
        int col = wn * 64 + j * 16 + l15;          // B frag: N = lane%16 (+tile)
        b[j].x = Bs[col][kk + kh];
        b[j].y = Bs[col][kk + kh + 1];
      }
#pragma unroll
      for (int i = 0; i < 2; ++i)
#pragma unroll
        for (int j = 0; j < 4; ++j)
          acc[i][j] = WMMA_F32(a[i], b[j], acc[i][j]);
    }
    __syncthreads();
  }

  // C/D layout: lanes0-15 N=lane, VGPR r -> M=r ; lanes16-31 N=lane-16, M=r+8
#pragma unroll
  for (int i = 0; i < 2; ++i)
#pragma unroll
    for (int j = 0; j < 4; ++j) {
      int n = n0 + wn * 64 + j * 16 + l15;
      float bv = bias ? bias[n] : 0.0f;
      int mb = m0 + wm * 32 + i * 16 + ((lane >> 4) << 3);
#pragma unroll
      for (int r = 0; r < 8; ++r)
        C[(size_t)(mb + r) * ldc + n] = alpha * acc[i][j][r] + bv;
    }
}

// Repack [B,S,D] -> per-head qh [B,H,S,64] and vb [B,H,S,128] (cols 0-63 = raw v).
__global__ void repack_heads(const float* __restrict__ mq, const float* __restrict__ mv,
                             float* __restrict__ qh, float* __restrict__ vb) {
  int idx = blockIdx.x * 256 + threadIdx.x;       // < B*S*D (exact multiple)
  int b = idx / (Ss * Dd);
  int r = idx % (Ss * Dd);
  int s = r / Dd;
  int c = r % Dd;
  int h = c / HDh, d = c % HDh;
  size_t row = (size_t)(b * Hh + h) * Ss + s;
  qh[row * HDh + d] = mq[idx];
  vb[row * 128 + d] = mv[idx];
}

// Seq-axis L2 norm: read vb cols 0-63, write normalized copy into cols 64-127.
__global__ __launch_bounds__(256)
void colnorm(float* __restrict__ vb) {
  int bc = blockIdx.x;                 // b*768 + c
  int b = bc / Dd, c = bc % Dd;
  int h = c / HDh, d = c % HDh;
  size_t base = ((size_t)(b * Hh + h) * Ss) * 128 + d;
  __shared__ float red[256];
  float s = 0.0f;
  for (int t = threadIdx.x; t < Ss; t += 256) {
    float v = vb[base + (size_t)t * 128];
    s += v * v;
  }
  red[threadIdx.x] = s;
  __syncthreads();
  for (int off = 128; off > 0; off >>= 1) {
    if (threadIdx.x < off) red[threadIdx.x] += red[threadIdx.x + off];
    __syncthreads();
  }
  float inv = 1.0f / fmaxf(sqrtf(red[0]), 1e-12f);
  for (int t = threadIdx.x; t < Ss; t += 256) {
    size_t o = base + (size_t)t * 128;
    vb[o + 64] = vb[o] * inv;
  }
}

// Row softmax over S=1024, in place. One block per row.
__global__ __launch_bounds__(256)
void softmax_rows(float* __restrict__ P) {
  size_t base = (size_t)blockIdx.x * Ss;
  __shared__ float red[256];
  float m = -3.402823e38f;
  for (int t = threadIdx.x; t < Ss; t += 256) m = fmaxf(m, P[base + t]);
  red[threadIdx.x] = m;
  __syncthreads();
  for (int off = 128; off > 0; off >>= 1) {
    if (threadIdx.x < off) red[threadIdx.x] = fmaxf(red[threadIdx.x], red[threadIdx.x + off]);
    __syncthreads();
  }
  m = red[0];
  __syncthreads();
  float sum = 0.0f;
  for (int t = threadIdx.x; t < Ss; t += 256) {
    float e = expf(P[base + t] - m);
    P[base + t] = e;
    sum += e;
  }
  red[threadIdx.x] = sum;
  __syncthreads();
  for (int off = 128; off > 0; off >>= 1) {
    if (threadIdx.x < off) red[threadIdx.x] += red[threadIdx.x + off];
    __syncthreads();
  }
  float inv = 1.0f / red[0];
  for (int t = threadIdx.x; t < Ss; t += 256) P[base + t] *= inv;
}

// In-place symmetrize: P = 0.5*(P + P^T). Thread (i,j) with j>=i writes both.
__global__ void symmetrize(float* __restrict__ P) {
  int idx = blockIdx.x * 256 + threadIdx.x;    // < S*S exact
  int i = idx >> 10, j = idx & 1023;
  if (j < i) return;
  float a = P[(size_t)i * Ss + j];
  float b = P[(size_t)j * Ss + i];
  float v = 0.5f * (a + b);
  P[(size_t)i * Ss + j] = v;
  P[(size_t)j * Ss + i] = v;
}

__global__ __launch_bounds__(256)
void rowsum(const float* __restrict__ P, float* __restrict__ dvec) {
  size_t base = (size_t)blockIdx.x * Ss;
  __shared__ float red[256];
  float s = 0.0f;
  for (int t = threadIdx.x; t < Ss; t += 256) s += P[base + t];
  red[threadIdx.x] = s;
  __syncthreads();
  for (int off = 128; off > 0; off >>= 1) {
    if (threadIdx.x < off) red[threadIdx.x] += red[threadIdx.x + off];
    __syncthreads();
  }
  if (threadIdx.x == 0) dvec[blockIdx.x] = red[0];
}

// Per-head epilogue: scatter ctx cols into [B,S,D]; accumulate tr = sum V*(V - (P@Vn)/dvec).
__global__ __launch_bounds__(256)
void head_epilogue(const float* __restrict__ ctxW, const float* __restrict__ vb_h,
                   const float* __restrict__ dvec, float* __restrict__ ctx_flat,
                   float* __restrict__ acc, int b, int h) {
  int idx = blockIdx.x * 256 + threadIdx.x;    // < S*64 exact
  int s = idx >> 6, d = idx & 63;
  float cv = ctxW[(size_t)s * 128 + d];
  ctx_flat[((size_t)b * Ss + s) * Dd + h * HDh + d] = cv;
  float lv = vb_h[(size_t)s * 128 + 64 + d];                 // normalized V
  float pnv = ctxW[(size_t)s * 128 + 64 + d] / dvec[s];      // (P@Vn)/rowsum
  float contrib = lv * (lv - pnv);
  __shared__ float red[256];
  red[threadIdx.x] = contrib;
  __syncthreads();
  for (int off = 128; off > 0; off >>= 1) {
    if (threadIdx.x < off) red[threadIdx.x] += red[threadIdx.x + off];
    __syncthreads();
  }
  if (threadIdx.x == 0) atomicAdd(&acc[0], red[0]);
}

// VtV = Vn^T Vn per head (64x64, K=1024) via WMMA; fused sum((VtV - I)^2).
// Vn lives in vb cols 64-127 (row stride 128).
__global__ __launch_bounds__(512)
void vtv_reg(const float* __restrict__ vb, float* __restrict__ acc) {
  const float* Vh = vb + (size_t)blockIdx.x * Ss * 128 + 64;
  int tid = threadIdx.x, lane = tid & 31, wid = tid >> 5;
  int tm = wid >> 2, tn = wid & 3;               // 4x4 grid of 16x16 tiles
  int l15 = lane & 15, kh = (lane >> 4) << 1;
  v8f c = {};
  for (int k0 = 0; k0 < Ss; k0 += 4) {
    v2f a, b;
    a.x = Vh[(size_t)(k0 + kh) * 128 + tm * 16 + l15];      // A = Vn^T: A[m][k]=Vn[k][m]
    a.y = Vh[(size_t)(k0 + kh + 1) * 128 + tm * 16 + l15];
    b.x = Vh[(size_t)(k0 + kh) * 128 + tn * 16 + l15];      // B = Vn
    b.y = Vh[(size_t)(k0 + kh + 1) * 128 + tn * 16 + l15];
    c = WMMA_F32(a, b, c);
  }
  float local = 0.0f;
  int n = tn * 16 + l15;
  int mb = tm * 16 + ((lane >> 4) << 3);
#pragma unroll
  for (int r = 0; r < 8; ++r) {
    float v = c[r] - ((mb + r) == n ? 1.0f : 0.0f);
    local += v * v;
  }
  __shared__ float red[512];
  red[tid] = local;
  __syncthreads();
  for (int off = 256; off > 0; off >>= 1) {
    if (tid < off) red[tid] += red[tid + off];
    __syncthreads();
  }
  if (tid == 0) atomicAdd(&acc[1], red[0]);
}

__global__ void zero_acc(float* acc) {
  if (threadIdx.x < 2) acc[threadIdx.x] = 0.0f;
}

__global__ void finalize(const float* __restrict__ acc, float* out_tr, float* out_reg) {
  *out_tr = fabsf(acc[0]) / 3072.0f;   // denom = HD*H*B = 64*12*4
  *out_reg = acc[1] / 3072.0f;
}

extern "C" void kernel_launch(void* const* d_in, const int* in_sizes, int n_in,
                              void* d_out, int out_size, void* d_ws, size_t ws_size,
                              hipStream_t stream) {
  (void)in_sizes; (void)n_in; (void)out_size; (void)ws_size;
  const float* hs = (const float*)d_in[0];
  const float* Wq = (const float*)d_in[1];
  const float* bq = (const float*)d_in[2];
  const float* Wv = (const float*)d_in[3];
  const float* bv = (const float*)d_in[4];
  const float* Wo = (const float*)d_in[5];
  const float* bo = (const float*)d_in[6];
  float* out = (float*)d_out;

  const size_t NE = (size_t)Bb * Ss * Dd;   // 3,145,728
  float* att     = out;                      // output 0
  float* out_tr  = out + NE;                 // output 1
  float* out_reg = out + NE + 1;             // output 2
  float* mq1     = out + NE + 2;             // output 3 (mixed_q)
  float* mq2     = mq1 + NE;                 // output 4 (mixed_q again)

  float* ws   = (float*)d_ws;                // ~65 MB of workspace used
  float* mv   = ws;                          // [B,S,D]
  float* qh   = mv + NE;                     // [B,H,S,64]
  float* vb   = qh + NE;                     // [B,H,S,128]: 0-63 raw v, 64-127 normalized V
  float* ctxf = vb + 2 * NE;                 // [B,S,D]
  float* P    = ctxf + NE;                   // one head's S x S probs (4 MB, L2-resident)
  float* ctxW = P + (size_t)Ss * Ss;         // [S,128]: cols 0-63 P@v, 64-127 P@Vn
  float* dvec = ctxW + (size_t)Ss * 128;     // [S]
  float* acc  = dvec + Ss;                   // {tr_sum, reg_sum}

  dim3 blk(128);

  // Projections (fp32 WMMA; mixed_q is an exact fp32 output). N=768 = 6 x 128.
  gemm_wmma_f32<<<dim3(Dd / 128, (Bb * Ss) / 64), blk, 0, stream>>>(
      hs, Wq, mq1, bq, Bb * Ss, Dd, Dd, Dd, Dd, Dd, 1.0f, 0, 0);
  gemm_wmma_f32<<<dim3(Dd / 128, (Bb * Ss) / 64), blk, 0, stream>>>(
      hs, Wv, mv, bv, Bb * Ss, Dd, Dd, Dd, Dd, Dd, 1.0f, 0, 0);
  hipMemcpyAsync(mq2, mq1, NE * sizeof(float), hipMemcpyDeviceToDevice, stream);

  repack_heads<<<(int)(NE / 256), 256, 0, stream>>>(mq1, mv, qh, vb);
  colnorm<<<Bb * Dd, 256, 0, stream>>>(vb);
  zero_acc<<<1, 32, 0, stream>>>(acc);
  vtv_reg<<<Bb * Hh, 512, 0, stream>>>(vb, acc);

  // Per-head pipeline keeps the 4 MB score matrix resident in L2.
  for (int bh = 0; bh < Bb * Hh; ++bh) {
    int b = bh / Hh, h = bh % Hh;
    const float* qh_h = qh + (size_t)bh * Ss * HDh;
    const float* vb_h = vb + (size_t)bh * Ss * 128;

    // scores = Q Q^T / 8   (B-operand transposed). N=1024 = 8 x 128.
    gemm_wmma_f32<<<dim3(Ss / 128, Ss / 64), blk, 0, stream>>>(
        qh_h, qh_h, P, nullptr, Ss, Ss, HDh, HDh, HDh, Ss, 0.125f, 0, 1);
    softmax_rows<<<Ss, 256, 0, stream>>>(P);
    symmetrize<<<(Ss * Ss) / 256, 256, 0, stream>>>(P);
    rowsum<<<Ss, 256, 0, stream>>>(P, dvec);

    // One N=128 GEMM: [P @ v | P @ Vn]  (A-tile of P loaded once for both)
    gemm_wmma_f32<<<dim3(1, Ss / 64), blk, 0, stream>>>(
        P, vb_h, ctxW, nullptr, Ss, 128, Ss, Ss, 128, 128, 1.0f, 0, 0);

    head_epilogue<<<(Ss * HDh) / 256, 256, 0, stream>>>(ctxW, vb_h, dvec, ctxf, acc, b, h);
  }

  // attention_output = ctx @ Wo + bo
  gemm_wmma_f32<<<dim3(Dd / 128, (Bb * Ss) / 64), blk, 0, stream>>>(
      ctxf, Wo, att, bo, Bb * Ss, Dd, Dd, Dd, Dd, Dd, 1.0f, 0, 0);

  finalize<<<1, 1, 0, stream>>>(acc, out_tr, out_reg);
}